// DomainGeneralisationBN_57415122813331
// MI455X (gfx1250) — compile-verified
//
#include <hip/hip_runtime.h>
#include <math.h>

// ---------------------------------------------------------------------------
// DomainGeneralisationBN on MI455X (gfx1250, wave32).
// One wave (32 lanes) owns one 32x32 SPD matrix.
//  - all 32x32 GEMMs through V_WMMA_F32_16X16X4_F32 (32 wmma per GEMM)
//  - eigh via parallel-ordered Jacobi: 16 disjoint rotations per round,
//    31 rounds per sweep; rotation params broadcast with readlane shuffles
//  - X tiles staged global->LDS with global_load_async_to_lds_b128,
//    double-buffered across the q loop (s_wait_asynccnt 8 / 0)
// ---------------------------------------------------------------------------

typedef float v2f __attribute__((ext_vector_type(2)));
typedef float v4f __attribute__((ext_vector_type(4)));
typedef float v8f __attribute__((ext_vector_type(8)));

#define NSWEEP 10
#define EPSV 1e-5f

// ---------------------------------------------------------------------------
// Async global -> LDS staging (ASYNCcnt-tracked, CDNA5 VGLOBAL async ops).
__device__ __forceinline__ unsigned lds_off_u32(const void* p) {
  return (unsigned)(unsigned long long)p;   // generic LDS addr: low 32 bits = LDS offset
}

// Copy one 32x32 f32 matrix (4KB): 32 lanes x 8 x B128 async transfers.
__device__ __forceinline__ void async_load_mat(float* dstLDS, const float* src, int lane) {
  unsigned base = lds_off_u32(dstLDS);
#pragma unroll
  for (int r = 0; r < 8; ++r) {
    unsigned loff = base + (unsigned)((r * 32 + lane) * 16);
    const void* g = (const char*)src + (size_t)(r * 32 + lane) * 16;
    asm volatile("global_load_async_to_lds_b128 %0, %1, off"
                 :: "v"(loff), "v"(g) : "memory");
  }
}
__device__ __forceinline__ void wait_async0() {
  asm volatile("s_wait_asynccnt 0x0" ::: "memory");
}
__device__ __forceinline__ void wait_async8() {   // previous 8-op matrix load done
  asm volatile("s_wait_asynccnt 0x8" ::: "memory");
}

// ---------------------------------------------------------------------------
// C(32x32) = op(A) * op(B). WMMA f32 16x16x4 layouts per ISA 7.12.2:
//   A[m][k]: lane = (k>>1)*16 + m, vgpr = k&1
//   B[k][n]: lane = (k>>1)*16 + n, vgpr = k&1
//   C/D    : vgpr r -> row = tile_m*16 + (lane>>4)*8 + r, col = tile_n*16 + (lane&15)
__device__ __forceinline__ void matmul32(const float* A, const float* B, float* C,
                                         int lane, bool transA, bool transB) {
  const int m15 = lane & 15;
  const int hi  = lane >> 4;
#pragma unroll
  for (int ti = 0; ti < 2; ++ti) {
#pragma unroll
    for (int tj = 0; tj < 2; ++tj) {
      v8f c = {0.f, 0.f, 0.f, 0.f, 0.f, 0.f, 0.f, 0.f};
#pragma unroll
      for (int kb = 0; kb < 8; ++kb) {
        const int k0 = kb * 4 + hi * 2;
        const int am = ti * 16 + m15;
        const int bn = tj * 16 + m15;
        v2f a, b;
        a.x = transA ? A[(k0    ) * 32 + am] : A[am * 32 + (k0    )];
        a.y = transA ? A[(k0 + 1) * 32 + am] : A[am * 32 + (k0 + 1)];
        b.x = transB ? B[bn * 32 + (k0    )] : B[(k0    ) * 32 + bn];
        b.y = transB ? B[bn * 32 + (k0 + 1)] : B[(k0 + 1) * 32 + bn];
        c = __builtin_amdgcn_wmma_f32_16x16x4_f32(false, a, false, b,
                                                  (short)0, c, false, false);
      }
      const int crow = ti * 16 + hi * 8;
      const int ccol = tj * 16 + m15;
#pragma unroll
      for (int r = 0; r < 8; ++r) C[(crow + r) * 32 + ccol] = c[r];
    }
  }
}

// ---------------------------------------------------------------------------
// Parallel-ordered Jacobi. Round-robin pairing for n=32: fix 31; in round r,
// pair j=0: (31, r); pair j>0: ((r+j) mod 31, (r-j) mod 31). 16 disjoint
// rotations per round -> A' = J^T A J with J a product of commuting plane
// rotations: all row updates, then all column (+V) updates. Wave-internal
// in-order DS semantics make this barrier-free.
__device__ __forceinline__ void pair_of(int r, int j, int& p, int& q) {
  if (j == 0) { p = 31; q = r; }
  else {
    int a = r + j;      if (a >= 31) a -= 31;
    int b = r + 31 - j; if (b >= 31) b -= 31;
    p = a; q = b;
  }
}

__device__ void jacobi32(float* A, float* V, int lane, bool wantV) {
  if (wantV) {
    for (int r = 0; r < 32; ++r) V[r * 32 + lane] = (r == lane) ? 1.f : 0.f;
  }
  const int j = lane & 15;   // lanes 16..31 duplicate lanes 0..15 (harmless)
  for (int sweep = 0; sweep < NSWEEP; ++sweep) {
    for (int r = 0; r < 31; ++r) {
      // --- per-lane rotation for its pair (reads before any store) ---
      int p, q; pair_of(r, j, p, q);
      float app = A[p * 33];
      float aqq = A[q * 33];
      float apq = A[p * 32 + q];
      float theta = 0.5f * (aqq - app) / apq;
      float t = copysignf(1.f, theta) / (fabsf(theta) + sqrtf(1.f + theta * theta));
      float cv = 1.f / sqrtf(1.f + t * t);
      float sv = t * cv;
      if (!(fabsf(apq) > 1e-9f * (fabsf(app) + fabsf(aqq)))) { cv = 1.f; sv = 0.f; }
      // --- row phase: B = J^T A (disjoint row pairs) ---
#pragma unroll
      for (int jj = 0; jj < 16; ++jj) {
        float cj = __shfl(cv, jj, 32);
        float sj = __shfl(sv, jj, 32);
        int pj, qj; pair_of(r, jj, pj, qj);
        float rp = A[pj * 32 + lane], rq = A[qj * 32 + lane];
        A[pj * 32 + lane] = cj * rp - sj * rq;
        A[qj * 32 + lane] = sj * rp + cj * rq;
      }
      // --- column phase: A' = B J ; V <- V J ---
#pragma unroll
      for (int jj = 0; jj < 16; ++jj) {
        float cj = __shfl(cv, jj, 32);
        float sj = __shfl(sv, jj, 32);
        int pj, qj; pair_of(r, jj, pj, qj);
        float cp = A[lane * 32 + pj], cq = A[lane * 32 + qj];
        A[lane * 32 + pj] = cj * cp - sj * cq;
        A[lane * 32 + qj] = sj * cp + cj * cq;
        if (wantV) {
          float vp = V[lane * 32 + pj], vq = V[lane * 32 + qj];
          V[lane * 32 + pj] = cj * vp - sj * vq;
          V[lane * 32 + qj] = sj * vp + cj * vq;
        }
      }
    }
  }
}

__device__ __forceinline__ float applyfn(float lam, int fn, float pexp) {
  switch (fn) {
    case 0:  return sqrtf(fmaxf(lam, 1e-30f));
    case 1:  return 1.f / sqrtf(fmaxf(lam, 1e-30f));
    case 2:  return logf(fmaxf(lam, 1e-30f));
    case 3:  return expf(lam);
    default: return expf(pexp * logf(fmaxf(lam, 1e-30f)));   // lam^p
  }
}

// Out = V * diag(f(lam)) * V^T, lam from diag(A). W is LDS scratch.
__device__ void matfn_recon(const float* A, const float* V, float* W, float* Out,
                            int lane, int fn, float pexp) {
  float f = applyfn(A[lane * 32 + lane], fn, pexp);
  for (int r = 0; r < 32; ++r) W[r * 32 + lane] = V[r * 32 + lane] * f;
  __syncthreads();
  matmul32(W, V, Out, lane, false, true);   // W * V^T
}

// ---------------------------------------------------------------------------

__global__ void k_zero(float* ws, int nfloats) {
  int i = blockIdx.x * blockDim.x + threadIdx.x;
  if (i < nfloats) ws[i] = 0.f;
}

__global__ void k_count(const int* __restrict__ ds, int* cnt, int N) {
  int i = blockIdx.x * blockDim.x + threadIdx.x;
  if (i < N) atomicAdd(&cnt[ds[i]], 1);
}

// Streaming reduction: per-n block sums its q matrices in B128-vector regs,
// one atomic pass into acc0[d].
__global__ void __launch_bounds__(32) k_accumG0(const float* __restrict__ X,
                                                const int* __restrict__ ds,
                                                float* acc0, int q) {
  int nidx = blockIdx.x, lane = threadIdx.x;
  int d = ds[nidx];
  v4f acc[8];
#pragma unroll
  for (int r = 0; r < 8; ++r) acc[r] = (v4f){0.f, 0.f, 0.f, 0.f};
  const v4f* base = (const v4f*)(X + (size_t)nidx * q * 1024);
  for (int qq = 0; qq < q; ++qq) {
    if (qq + 1 < q) __builtin_prefetch(base + (size_t)(qq + 1) * 256);
    const v4f* Xm = base + (size_t)qq * 256;
#pragma unroll
    for (int r = 0; r < 8; ++r) acc[r] += Xm[r * 32 + lane];
  }
  float* A = acc0 + (size_t)d * 1024;
#pragma unroll
  for (int r = 0; r < 8; ++r) {
    int idx = (r * 32 + lane) * 4;
    unsafeAtomicAdd(&A[idx + 0], acc[r].x);
    unsafeAtomicAdd(&A[idx + 1], acc[r].y);
    unsafeAtomicAdd(&A[idx + 2], acc[r].z);
    unsafeAtomicAdd(&A[idx + 3], acc[r].w);
  }
}

// G0 = acc0/count -> G0^{1/2}, G0^{-1/2}; also B^{1/2}.
__global__ void __launch_bounds__(32) k_domain1(const float* acc0, const float* Bmat,
                                                const int* cnt, int q,
                                                float* g0sq, float* g0isq, float* bsq) {
  __shared__ float sA[1024], sV[1024], sW[1024];
  int d = blockIdx.x, lane = threadIdx.x;
  int c = cnt[d]; if (c < 1) c = 1;
  float w = 1.f / (float)(c * q);
  for (int r = 0; r < 32; ++r) sA[r * 32 + lane] = acc0[d * 1024 + r * 32 + lane] * w;
  __syncthreads();
  jacobi32(sA, sV, lane, true);
  __syncthreads();
  matfn_recon(sA, sV, sW, g0sq + d * 1024, lane, 0, 0.f);
  __syncthreads();
  matfn_recon(sA, sV, sW, g0isq + d * 1024, lane, 1, 0.f);
  __syncthreads();
  for (int r = 0; r < 32; ++r) sA[r * 32 + lane] = Bmat[d * 1024 + r * 32 + lane];
  __syncthreads();
  jacobi32(sA, sV, lane, true);
  __syncthreads();
  matfn_recon(sA, sV, sW, bsq + d * 1024, lane, 0, 0.f);
}

// XT = Gs * logm(Gi X Gi) * Gs accumulated (weighted) into accT[d].
// X staged with double-buffered async B128 copies.
__global__ void __launch_bounds__(32) k_XT(const float* __restrict__ X,
                                           const int* __restrict__ ds,
                                           const float* g0sq, const float* g0isq,
                                           const int* cnt, int q, float* accT) {
  __shared__ float sX0[1024], sX1[1024], sA[1024], sV[1024], sW[1024], sT[1024], sAcc[1024];
  int nidx = blockIdx.x, lane = threadIdx.x;
  int d = ds[nidx];
  const float* Gs = g0sq + d * 1024;
  const float* Gi = g0isq + d * 1024;
  for (int r = 0; r < 32; ++r) sAcc[r * 32 + lane] = 0.f;
  const float* base = X + (size_t)nidx * q * 1024;
  async_load_mat(sX0, base, lane);
  for (int qq = 0; qq < q; ++qq) {
    float* sX   = (qq & 1) ? sX1 : sX0;
    float* sNxt = (qq & 1) ? sX0 : sX1;
    if (qq + 1 < q) { async_load_mat(sNxt, base + (size_t)(qq + 1) * 1024, lane); wait_async8(); }
    else            { wait_async0(); }
    __syncthreads();
    matmul32(Gi, sX, sT, lane, false, false);      // Gi * X
    __syncthreads();
    matmul32(sT, Gi, sA, lane, false, false);      // inner = Gi X Gi
    __syncthreads();
    jacobi32(sA, sV, lane, true);
    __syncthreads();
    matfn_recon(sA, sV, sW, sX, lane, 2, 0.f);     // sX = logm(inner)
    __syncthreads();
    matmul32(Gs, sX, sT, lane, false, false);
    __syncthreads();
    matmul32(sT, Gs, sX, lane, false, false);      // XT
    __syncthreads();
    for (int r = 0; r < 32; ++r) sAcc[r * 32 + lane] += sX[r * 32 + lane];
    __syncthreads();
  }
  int c = cnt[d]; if (c < 1) c = 1;
  float w = 1.f / (float)(c * q);
  float* A = accT + (size_t)d * 1024;
  for (int r = 0; r < 32; ++r) unsafeAtomicAdd(&A[r * 32 + lane], sAcc[r * 32 + lane] * w);
}

// G = G0sq * expm(G0isq GT G0isq) * G0sq;  gisq = G^{-1/2}.
__global__ void __launch_bounds__(32) k_domain2(const float* accT, const float* g0sq,
                                                const float* g0isq, float* gisq) {
  __shared__ float sA[1024], sV[1024], sW[1024], sT1[1024], sT2[1024];
  int d = blockIdx.x, lane = threadIdx.x;
  const float* Gs = g0sq + d * 1024;
  const float* Gi = g0isq + d * 1024;
  const float* GT = accT + d * 1024;
  matmul32(Gi, GT, sT1, lane, false, false);
  __syncthreads();
  matmul32(sT1, Gi, sA, lane, false, false);
  __syncthreads();
  jacobi32(sA, sV, lane, true);
  __syncthreads();
  matfn_recon(sA, sV, sW, sT1, lane, 3, 0.f);      // expm
  __syncthreads();
  matmul32(Gs, sT1, sT2, lane, false, false);
  __syncthreads();
  matmul32(sT2, Gs, sA, lane, false, false);       // G
  __syncthreads();
  jacobi32(sA, sV, lane, true);
  __syncthreads();
  matfn_recon(sA, sV, sW, gisq + d * 1024, lane, 1, 0.f);
}

// var[d] += w * sum_i log^2(lambda_i(Xc)); eigenvalues only (no V updates).
__global__ void __launch_bounds__(32) k_var(const float* __restrict__ X,
                                            const int* __restrict__ ds,
                                            const float* gisq, const int* cnt,
                                            int q, float* var) {
  __shared__ float sX0[1024], sX1[1024], sA[1024], sT[1024];
  int nidx = blockIdx.x, lane = threadIdx.x;
  int d = ds[nidx];
  const float* Gi = gisq + d * 1024;
  float acc = 0.f;
  const float* base = X + (size_t)nidx * q * 1024;
  async_load_mat(sX0, base, lane);
  for (int qq = 0; qq < q; ++qq) {
    float* sX   = (qq & 1) ? sX1 : sX0;
    float* sNxt = (qq & 1) ? sX0 : sX1;
    if (qq + 1 < q) { async_load_mat(sNxt, base + (size_t)(qq + 1) * 1024, lane); wait_async8(); }
    else            { wait_async0(); }
    __syncthreads();
    matmul32(Gi, sX, sT, lane, false, false);
    __syncthreads();
    matmul32(sT, Gi, sA, lane, false, false);      // Xc
    __syncthreads();
    jacobi32(sA, (float*)0, lane, false);
    __syncthreads();
    float l = logf(fmaxf(sA[lane * 32 + lane], 1e-30f));
    acc += l * l;
    __syncthreads();
  }
  for (int off = 16; off > 0; off >>= 1) acc += __shfl_down(acc, off, 32);
  if (lane == 0) {
    int c = cnt[d]; if (c < 1) c = 1;
    unsafeAtomicAdd(&var[d], acc / (float)(c * q));
  }
}

__global__ void k_p(const float* var, float* p, int D) {
  int i = threadIdx.x;
  if (i < D) p[i] = sqrtf(1.f / (var[i] + EPSV));
}

// Xc = Gi X Gi -> Xp = V lam^p V^T -> Xr = R Xp R^T -> Xb = Bsq Xr Bsq.
__global__ void __launch_bounds__(32) k_final(const float* __restrict__ X,
                                              const int* __restrict__ ds,
                                              const float* gisq, const float* R,
                                              const float* bsq, const float* pvec,
                                              int q, float* __restrict__ out) {
  __shared__ float sX[1024], sA[1024], sV[1024], sW[1024], sT[1024];
  int m = blockIdx.x, lane = threadIdx.x;
  int d = ds[m / q];
  const float* Gi = gisq + d * 1024;
  const float* Rd = R + d * 1024;
  const float* Bs = bsq + d * 1024;
  float pe = pvec[d];
  async_load_mat(sX, X + (size_t)m * 1024, lane);
  wait_async0();
  __syncthreads();
  matmul32(Gi, sX, sT, lane, false, false);
  __syncthreads();
  matmul32(sT, Gi, sA, lane, false, false);        // Xc
  __syncthreads();
  jacobi32(sA, sV, lane, true);
  __syncthreads();
  matfn_recon(sA, sV, sW, sX, lane, 4, pe);        // Xp = PowerEig(Xc, p)
  __syncthreads();
  matmul32(Rd, sX, sT, lane, false, false);        // R * Xp
  __syncthreads();
  matmul32(sT, Rd, sX, lane, false, true);         // ... * R^T
  __syncthreads();
  matmul32(Bs, sX, sT, lane, false, false);
  __syncthreads();
  matmul32(sT, Bs, sX, lane, false, false);        // Xb
  __syncthreads();
  v4f* Om = (v4f*)(out + (size_t)m * 1024);
  const v4f* sXv = (const v4f*)sX;
#pragma unroll
  for (int r = 0; r < 8; ++r) Om[r * 32 + lane] = sXv[r * 32 + lane];
}

// ---------------------------------------------------------------------------

extern "C" void kernel_launch(void* const* d_in, const int* in_sizes, int n_in,
                              void* d_out, int out_size, void* d_ws, size_t ws_size,
                              hipStream_t stream) {
  const float* X  = (const float*)d_in[0];
  const int*   ds = (const int*)  d_in[1];
  const float* R  = (const float*)d_in[2];
  const float* B  = (const float*)d_in[3];
  float* out = (float*)d_out;

  const int N = in_sizes[1];
  const int q = in_sizes[0] / (N * 1024);
  const int D = in_sizes[2] / 1024;
  const int M = N * q;

  float* ws    = (float*)d_ws;
  float* acc0  = ws;
  float* accT  = acc0  + (size_t)D * 1024;
  float* g0sq  = accT  + (size_t)D * 1024;
  float* g0isq = g0sq  + (size_t)D * 1024;
  float* gisq  = g0isq + (size_t)D * 1024;
  float* bsq   = gisq  + (size_t)D * 1024;
  float* var   = bsq   + (size_t)D * 1024;
  float* pvec  = var + D;
  int*   cnt   = (int*)(pvec + D);

  const int zf = D * 1024 * 6 + 3 * D;   // accumulators + var + p + cnt (0.0f == 0 bits)
  k_zero   <<<(zf + 255) / 256, 256, 0, stream>>>(ws, zf);
  k_count  <<<(N + 255) / 256, 256, 0, stream>>>(ds, cnt, N);
  k_accumG0<<<N, 32, 0, stream>>>(X, ds, acc0, q);
  k_domain1<<<D, 32, 0, stream>>>(acc0, B, cnt, q, g0sq, g0isq, bsq);
  k_XT     <<<N, 32, 0, stream>>>(X, ds, g0sq, g0isq, cnt, q, accT);
  k_domain2<<<D, 32, 0, stream>>>(accT, g0sq, g0isq, gisq);
  k_var    <<<N, 32, 0, stream>>>(X, ds, gisq, cnt, q, var);
  k_p      <<<1, 32, 0, stream>>>(var, pvec, D);
  k_final  <<<M, 32, 0, stream>>>(X, ds, gisq, R, bsq, pvec, q, out);
}